// SSMKernelDiag_85830626443731
// MI455X (gfx1250) — compile-verified
//
#include <hip/hip_runtime.h>

// S4D diagonal SSM kernel for MI455X (gfx1250).
// k[h, 64q+r] = 2*Re( sum_n w[h,n] * (dA^64)^q * dA^r )  -> per-head real GEMM
//   [Aq_re | Aq_imN] (64x128)  x  [Br_re ; Br_im] (128x64),  Aq_imN = -Aq_im
// executed with V_WMMA_F32_16X16X4_F32 (full fp32 precision).

typedef __attribute__((ext_vector_type(2))) float v2f;
typedef __attribute__((ext_vector_type(8))) float v8f;

#define H_DIM 512
#define N_DIM 64
#define L_DIM 4096
#define QQ 64           // l = 64*q + r
#define RR 64
#define PITCH 65        // LDS row padding

__global__ __launch_bounds__(512) void s4d_wmma_kernel(
    const float* __restrict__ log_step,  // (H)
    const float* __restrict__ A_real,    // (H,N)
    const float* __restrict__ A_imag,    // (H,N)
    const float* __restrict__ B_real,    // (H,N)
    const float* __restrict__ B_imag,    // (H,N)
    const float* __restrict__ C,         // (H,N,2)
    float* __restrict__ out)             // (H,L)
{
    __shared__ float sAre[QQ][PITCH];    // Aq_re[q][n]  =  Re(w * dA64^q)
    __shared__ float sAimN[QQ][PITCH];   // Aq_imN[q][n] = -Im(w * dA64^q)
    __shared__ float sBre[N_DIM][PITCH]; // Br_re[n][r]  =  Re(dA^r)
    __shared__ float sBim[N_DIM][PITCH]; // Br_im[n][r]  =  Im(dA^r)
    __shared__ float s_x[N_DIM], s_y[N_DIM], s_wre[N_DIM], s_wim[N_DIM];

    const int h   = blockIdx.x;
    const int tid = threadIdx.x;

    // ---- Phase 1: per-n complex scalars -------------------------------
    if (tid < N_DIM) {
        const int n = tid;
        const float dt = __expf(log_step[h]);
        const float ar = -__expf(A_real[h * N_DIM + n]);   // Re(A) < 0
        const float ai = A_imag[h * N_DIM + n];
        const float x  = dt * ar;                          // Re(dt*A)
        const float y  = dt * ai;                          // Im(dt*A)
        // dA = exp(dt*A)
        float sn, cs;
        const float mag = __expf(x);
        __sincosf(y, &sn, &cs);
        const float dAre = mag * cs, dAim = mag * sn;
        // g = (dA - 1)/A = (dA-1)*conj(A)/|A|^2
        const float nr = dAre - 1.0f, ni = dAim;
        const float inv = 1.0f / (ar * ar + ai * ai);
        const float gr = (nr * ar + ni * ai) * inv;
        const float gi = (ni * ar - nr * ai) * inv;
        // dB = dt * B * g
        const float br = B_real[h * N_DIM + n], bi = B_imag[h * N_DIM + n];
        const float dBre = dt * (br * gr - bi * gi);
        const float dBim = dt * (br * gi + bi * gr);
        // w = Cc * dB
        const float cr = C[(h * N_DIM + n) * 2 + 0];
        const float ci = C[(h * N_DIM + n) * 2 + 1];
        s_wre[n] = cr * dBre - ci * dBim;
        s_wim[n] = cr * dBim + ci * dBre;
        s_x[n] = x;
        s_y[n] = y;
    }
    __syncthreads();

    // ---- Phase 2: build GEMM operands in LDS --------------------------
    // Br[n][r] = dA^r = exp(r*(x+iy))   (direct exponentiation: no error chain)
    for (int i = tid; i < N_DIM * RR; i += 512) {
        const int n = i >> 6, r = i & 63;
        const float fr = (float)r;
        float sn, cs;
        const float mag = __expf(fr * s_x[n]);
        __sincosf(fr * s_y[n], &sn, &cs);
        sBre[n][r] = mag * cs;
        sBim[n][r] = mag * sn;
    }
    // Aq[q][n] = w[n] * dA^(64q);  imaginary panel stored NEGATED so the
    // GEMM loop needs no per-fragment sign flips (f32 WMMA has no A-neg).
    for (int i = tid; i < QQ * N_DIM; i += 512) {
        const int q = i >> 6, n = i & 63;
        const float t = 64.0f * (float)q;
        float sn, cs;
        const float mag = __expf(t * s_x[n]);
        __sincosf(t * s_y[n], &sn, &cs);
        const float pr = mag * cs, pi = mag * sn;
        sAre[q][n]  = s_wre[n] * pr - s_wim[n] * pi;
        sAimN[q][n] = -(s_wre[n] * pi) - s_wim[n] * pr;
    }
    __syncthreads();

    // ---- Phase 3: 64x64x128 real GEMM via V_WMMA_F32_16X16X4_F32 ------
    // out2Re[q][r] = sum_n sAre[q][n]*sBre[n][r] + sAimN[q][n]*sBim[n][r]
    const int wave = tid >> 5;     // 0..15 -> one 16x16 tile each
    const int lane = tid & 31;
    const int qt   = wave >> 2;    // tile row (q)
    const int rt   = wave & 3;     // tile col (r)
    const int lhi  = lane >> 4;    // 0/1: which K half the lane holds
    const int llo  = lane & 15;    // M (for A) / N (for B)

    v8f acc = {};
    const int m  = qt * 16 + llo;  // A row
    const int rc = rt * 16 + llo;  // B col
    for (int kk = 0; kk < 128; kk += 4) {
        v2f a, b;
#pragma unroll
        for (int j = 0; j < 2; ++j) {
            // ISA 32-bit A 16x4 layout: lane = M + 16*(K/2), vgpr j = K%2
            const int k = kk + 2 * lhi + j;
            if (k < 64) {               // real-part half of K
                a[j] = sAre[m][k];
                b[j] = sBre[k][rc];
            } else {                    // imaginary half (A pre-negated)
                a[j] = sAimN[m][k - 64];
                b[j] = sBim[k - 64][rc];
            }
        }
        // 8 args: (neg_a, A, neg_b, B, c_mod, C, reuse_a, reuse_b)
        acc = __builtin_amdgcn_wmma_f32_16x16x4_f32(
            false, a, false, b, (short)0, acc, false, false);
    }

    // D layout: VGPR v, lanes 0-15 -> M=v, lanes 16-31 -> M=v+8; N=llo
    float* outh = out + (size_t)h * L_DIM;
#pragma unroll
    for (int v = 0; v < 8; ++v) {
        const int M = v + 8 * lhi;
        const int l = 64 * (qt * 16 + M) + rt * 16 + llo;
        outh[l] = 2.0f * acc[v];
    }
}

extern "C" void kernel_launch(void* const* d_in, const int* in_sizes, int n_in,
                              void* d_out, int out_size, void* d_ws, size_t ws_size,
                              hipStream_t stream) {
    (void)in_sizes; (void)n_in; (void)d_ws; (void)ws_size; (void)out_size;
    const float* log_step = (const float*)d_in[0];
    const float* A_real   = (const float*)d_in[1];
    const float* A_imag   = (const float*)d_in[2];
    const float* B_real   = (const float*)d_in[3];
    const float* B_imag   = (const float*)d_in[4];
    const float* C        = (const float*)d_in[5];
    // d_in[6] is L (=4096) on device; shapes are compile-time constants here.
    float* out = (float*)d_out;

    s4d_wmma_kernel<<<H_DIM, 512, 0, stream>>>(log_step, A_real, A_imag,
                                               B_real, B_imag, C, out);
}